// ParallelMultiHeadAttention_87290915324639
// MI455X (gfx1250) — compile-verified
//
#include <hip/hip_runtime.h>
#include <hip/hip_bf16.h>

// MI455X / gfx1250 fp32 causal multi-head attention, fully on the WMMA path.
// All matmuls use V_WMMA_F32_16X16X4_F32 (native fp32 matrix op) to match the
// fp32 reference exactly. Main loops are software-pipelined with LDS double
// buffering (global loads for chunk k+1 overlap WMMA compute on chunk k).
// Workspace usage: 64MB (K,Q,V,Opre each 16MB).

typedef __attribute__((ext_vector_type(2))) float v2f;
typedef __attribute__((ext_vector_type(8))) float v8f;

#define T_SEQ 2048
#define C_DIM 1024
#define H_NUM 16
#define CH    64

__device__ __forceinline__ v8f wmma4(v2f a, v2f b, v8f c) {
  // D = A(16x4 f32) * B(4x16 f32) + C(16x16 f32)
  return __builtin_amdgcn_wmma_f32_16x16x4_f32(false, a, false, b,
                                               (short)0, c, false, false);
}

// ---------------------------------------------------------------------------
// Generic fp32 WMMA GEMM: block tile 128(M) x 64(N), K-chunks of 16,
// double-buffered LDS pipeline. 8 waves arranged 4(M) x 2(N); each wave owns
// a 32x32 subtile (2x2 WMMA tiles).
// MODE 0: epilogue scatters packed (h,c,s) columns into K/Q/V (B,H,T,Ch).
// MODE 1: epilogue adds bias and stores row-major.
// ---------------------------------------------------------------------------
template <int MODE>
__global__ __launch_bounds__(256)
void gemm_wmma(const float* __restrict__ A, const float* __restrict__ Bm,
               const float* __restrict__ bias,
               float* __restrict__ out0, float* __restrict__ out1,
               float* __restrict__ out2, int M, int N, int K)
{
  __shared__ float sa[2][128 * 18];  // stride 18: rows stay 8B aligned
  __shared__ float sb[2][16 * 66];   // stride 66: breaks bank stride-64

  const int tid  = threadIdx.x;
  const int lane = tid & 31;
  const int wv   = tid >> 5;
  const int lm   = lane & 15;
  const int lh   = lane >> 4;
  const int wm   = wv >> 1;          // 0..3
  const int wn   = wv & 1;           // 0..1
  const int M0   = blockIdx.y * 128;
  const int N0   = blockIdx.x * 64;

  // per-thread staging coordinates (constant across chunks)
  int ra_r[8], ra_c[8], rb_r[4], rb_c[4];
#pragma unroll
  for (int i = 0; i < 8; ++i) { int idx = i * 256 + tid; ra_r[i] = idx >> 4; ra_c[i] = idx & 15; }
#pragma unroll
  for (int i = 0; i < 4; ++i) { int idx = i * 256 + tid; rb_r[i] = idx >> 6; rb_c[i] = idx & 63; }

  const v8f vzero = {0.f, 0.f, 0.f, 0.f, 0.f, 0.f, 0.f, 0.f};
  v8f acc[2][2];
  acc[0][0] = vzero; acc[0][1] = vzero; acc[1][0] = vzero; acc[1][1] = vzero;

  const int nkb = K >> 4;
  float ga[8], gb[4];

  // prologue: stage chunk 0
#pragma unroll
  for (int i = 0; i < 8; ++i)
    ga[i] = A[(size_t)(M0 + ra_r[i]) * K + ra_c[i]];
#pragma unroll
  for (int i = 0; i < 4; ++i)
    gb[i] = Bm[(size_t)rb_r[i] * N + N0 + rb_c[i]];
#pragma unroll
  for (int i = 0; i < 8; ++i) sa[0][ra_r[i] * 18 + ra_c[i]] = ga[i];
#pragma unroll
  for (int i = 0; i < 4; ++i) sb[0][rb_r[i] * 66 + rb_c[i]] = gb[i];
  __syncthreads();

  for (int kb = 0; kb < nkb; ++kb) {
    const int cur = kb & 1;
    const bool hasNext = (kb + 1) < nkb;
    if (hasNext) {               // issue next chunk's global loads early
      const int k1 = (kb + 1) * 16;
#pragma unroll
      for (int i = 0; i < 8; ++i)
        ga[i] = A[(size_t)(M0 + ra_r[i]) * K + k1 + ra_c[i]];
#pragma unroll
      for (int i = 0; i < 4; ++i)
        gb[i] = Bm[(size_t)(k1 + rb_r[i]) * N + N0 + rb_c[i]];
    }

    const float* sac = sa[cur];
    const float* sbc = sb[cur];
#pragma unroll
    for (int ks = 0; ks < 4; ++ks) {
      const int kc = ks * 4;
      v2f a0 = *(const v2f*)&sac[(wm * 32 +      lm) * 18 + kc + 2 * lh];
      v2f a1 = *(const v2f*)&sac[(wm * 32 + 16 + lm) * 18 + kc + 2 * lh];
      v2f b0, b1;
      b0.x = sbc[(kc + 2 * lh    ) * 66 + wn * 32 +      lm];
      b0.y = sbc[(kc + 2 * lh + 1) * 66 + wn * 32 +      lm];
      b1.x = sbc[(kc + 2 * lh    ) * 66 + wn * 32 + 16 + lm];
      b1.y = sbc[(kc + 2 * lh + 1) * 66 + wn * 32 + 16 + lm];
      acc[0][0] = wmma4(a0, b0, acc[0][0]);
      acc[0][1] = wmma4(a0, b1, acc[0][1]);
      acc[1][0] = wmma4(a1, b0, acc[1][0]);
      acc[1][1] = wmma4(a1, b1, acc[1][1]);
    }

    if (hasNext) {               // fill the other buffer for the next round
      const int nxt = cur ^ 1;
#pragma unroll
      for (int i = 0; i < 8; ++i) sa[nxt][ra_r[i] * 18 + ra_c[i]] = ga[i];
#pragma unroll
      for (int i = 0; i < 4; ++i) sb[nxt][rb_r[i] * 66 + rb_c[i]] = gb[i];
    }
    __syncthreads();
  }

#pragma unroll
  for (int mi = 0; mi < 2; ++mi) {
#pragma unroll
    for (int ni = 0; ni < 2; ++ni) {
      const int n = N0 + wn * 32 + ni * 16 + lm;
      if (MODE == 0) {
        // column n = h*192 + c*3 + s   (s: 0=K, 1=Q, 2=V)
        const int h = n / 192;
        const int rem = n - h * 192;
        const int c = rem / 3;
        const int s = rem - c * 3;
        float* dst = (s == 0) ? out0 : (s == 1) ? out1 : out2;
#pragma unroll
        for (int v = 0; v < 8; ++v) {
          const int row = M0 + wm * 32 + mi * 16 + v + 8 * lh;  // b*T + t
          const int b = row >> 11, t = row & (T_SEQ - 1);
          dst[(((size_t)b * H_NUM + h) * T_SEQ + t) * CH + c] = acc[mi][ni][v];
        }
      } else {
        const float bv = bias[n];
#pragma unroll
        for (int v = 0; v < 8; ++v) {
          const int row = M0 + wm * 32 + mi * 16 + v + 8 * lh;
          out0[(size_t)row * N + n] = acc[mi][ni][v] + bv;
        }
      }
    }
  }
}

// ---------------------------------------------------------------------------
// Flash attention over transposed tiles, double-buffered K/V staging.
// Block = 8 waves = 128 query rows of one (b,h). Wave w owns queries
// q0 = band*128 + w*16. Tiles: S^T = K*Q^T (keys x queries), O^T = V^T * P
// (channels x queries), so each lane owns one query column and softmax
// statistics are per-lane (+ one shfl_xor 16 for the lane-half partner).
// ---------------------------------------------------------------------------
__global__ __launch_bounds__(256)
void attn_wmma(const float* __restrict__ Qb, const float* __restrict__ Kb,
               const float* __restrict__ Vb, float* __restrict__ Opre)
{
  __shared__ float kl[2][16 * 66];
  __shared__ float vl[2][16 * 66];
  __shared__ float pl[8 * 16 * 17];   // wave-private P round-trip

  const int tid  = threadIdx.x;
  const int lane = tid & 31;
  const int wv   = tid >> 5;
  const int lm   = lane & 15;
  const int lh   = lane >> 4;
  const int bh   = blockIdx.x >> 4;   // b*H + h  (matches Q/K/V layout)
  const int band = blockIdx.x & 15;
  const int b    = bh >> 4;
  const int h    = bh & 15;
  const int q0   = band * 128 + wv * 16;
  const float scale = 0.125f;         // Ch^-0.5 = 1/8

  const float* Qh = Qb + (size_t)bh * T_SEQ * CH;
  const float* Kh = Kb + (size_t)bh * T_SEQ * CH;
  const float* Vh = Vb + (size_t)bh * T_SEQ * CH;

  // per-thread staging coordinates
  int st_r[4], st_c[4];
#pragma unroll
  for (int i = 0; i < 4; ++i) { int idx = i * 256 + tid; st_r[i] = idx >> 6; st_c[i] = idx & 63; }

  // Q^T B-fragments held in registers for the whole key loop:
  // chunk kk covers channels 4kk..4kk+3; lane holds Q[q0+lm][4kk+2lh+{0,1}]
  v2f qf[16];
#pragma unroll
  for (int kk = 0; kk < 16; ++kk)
    qf[kk] = *(const v2f*)&Qh[(size_t)(q0 + lm) * CH + 4 * kk + 2 * lh];

  const v8f vzero = {0.f, 0.f, 0.f, 0.f, 0.f, 0.f, 0.f, 0.f};
  v8f o[4];                           // O^T tiles: channel groups 0,16,32,48
  o[0] = vzero; o[1] = vzero; o[2] = vzero; o[3] = vzero;
  float mrow = -__builtin_inff();
  float lrow = 0.f;

  const int jend = band * 8 + 8;      // key blocks staged for whole band
  const int jlim = band * 8 + wv;     // causal limit for this wave
  float* pw = pl + wv * 16 * 17;

  float gk[4], gv[4];
  // prologue: stage key/value tile j=0
#pragma unroll
  for (int i = 0; i < 4; ++i) {
    gk[i] = Kh[(size_t)st_r[i] * CH + st_c[i]];
    gv[i] = Vh[(size_t)st_r[i] * CH + st_c[i]];
  }
#pragma unroll
  for (int i = 0; i < 4; ++i) {
    kl[0][st_r[i] * 66 + st_c[i]] = gk[i];
    vl[0][st_r[i] * 66 + st_c[i]] = gv[i];
  }
  __syncthreads();

  for (int j = 0; j < jend; ++j) {
    const int cur = j & 1;
    const bool hasNext = (j + 1) < jend;
    if (hasNext) {                    // issue next tile's global loads early
      const int r1 = (j + 1) * 16;
#pragma unroll
      for (int i = 0; i < 4; ++i) {
        gk[i] = Kh[(size_t)(r1 + st_r[i]) * CH + st_c[i]];
        gv[i] = Vh[(size_t)(r1 + st_r[i]) * CH + st_c[i]];
      }
    }

    if (j <= jlim) {                  // wave-uniform: EXEC stays all-ones
      const float* klc = kl[cur];
      const float* vlc = vl[cur];
      // S^T = K * Q^T   (16 keys x 16 queries), contraction over Ch=64
      v8f s = vzero;
#pragma unroll
      for (int kk = 0; kk < 16; ++kk) {
        v2f a = *(const v2f*)&klc[lm * 66 + 4 * kk + 2 * lh];
        s = wmma4(a, qf[kk], s);
      }
      // scale + causal mask + tile max (per query column == per lane)
      const int qg = q0 + lm;
      float tmax = -__builtin_inff();
#pragma unroll
      for (int v = 0; v < 8; ++v) {
        const int kg = j * 16 + v + 8 * lh;
        float sv = s[v] * scale;
        sv = (kg > qg) ? -__builtin_inff() : sv;
        s[v] = sv;
        tmax = fmaxf(tmax, sv);
      }
      tmax = fmaxf(tmax, __shfl_xor(tmax, 16, 32));
      const float mnew  = fmaxf(mrow, tmax);
      const float alpha = __expf(mrow - mnew);
      float tsum = 0.f;
#pragma unroll
      for (int v = 0; v < 8; ++v) {
        const float pv = __expf(s[v] - mnew);
        s[v] = pv;
        tsum += pv;
      }
      tsum += __shfl_xor(tsum, 16, 32);
      lrow = lrow * alpha + tsum;
      mrow = mnew;
      o[0] *= alpha; o[1] *= alpha; o[2] *= alpha; o[3] *= alpha;

      // D-layout P -> LDS -> B-fragments (keys x queries)
#pragma unroll
      for (int v = 0; v < 8; ++v)
        pw[(v + 8 * lh) * 17 + lm] = s[v];
      v2f pf[4];
#pragma unroll
      for (int k2 = 0; k2 < 4; ++k2) {
        pf[k2].x = pw[(4 * k2 + 2 * lh    ) * 17 + lm];
        pf[k2].y = pw[(4 * k2 + 2 * lh + 1) * 17 + lm];
      }
      // O^T += V^T * P : A-frag = V^T (channels x keys) from LDS V tile
#pragma unroll
      for (int ct = 0; ct < 4; ++ct) {
#pragma unroll
        for (int k2 = 0; k2 < 4; ++k2) {
          v2f a;
          a.x = vlc[(4 * k2 + 2 * lh    ) * 66 + ct * 16 + lm];
          a.y = vlc[(4 * k2 + 2 * lh + 1) * 66 + ct * 16 + lm];
          o[ct] = wmma4(a, pf[k2], o[ct]);
        }
      }
    }

    if (hasNext) {                    // fill the other buffer
      const int nxt = cur ^ 1;
#pragma unroll
      for (int i = 0; i < 4; ++i) {
        kl[nxt][st_r[i] * 66 + st_c[i]] = gk[i];
        vl[nxt][st_r[i] * 66 + st_c[i]] = gv[i];
      }
    }
    __syncthreads();
  }

  // normalize and store with the reference's channel-major permute:
  // out channel index = c*H + h
  const float rinv = 1.f / lrow;
  const int tq = q0 + lm;
  float* Ob = Opre + ((size_t)b * T_SEQ + tq) * C_DIM + h;
#pragma unroll
  for (int ct = 0; ct < 4; ++ct) {
#pragma unroll
    for (int v = 0; v < 8; ++v) {
      const int ch = ct * 16 + v + 8 * lh;
      Ob[(size_t)ch * H_NUM] = o[ct][v] * rinv;
    }
  }
}

// ---------------------------------------------------------------------------
extern "C" void kernel_launch(void* const* d_in, const int* in_sizes, int n_in,
                              void* d_out, int out_size, void* d_ws, size_t ws_size,
                              hipStream_t stream) {
  (void)in_sizes; (void)n_in; (void)out_size; (void)ws_size;
  const float* x  = (const float*)d_in[0];  // (B,T,C)
  const float* Wu = (const float*)d_in[1];  // (C,3C)
  const float* Wp = (const float*)d_in[2];  // (C,C)
  const float* bp = (const float*)d_in[3];  // (C,)
  float* out = (float*)d_out;               // (B,T,C)

  const size_t SZ = (size_t)2 * H_NUM * T_SEQ * CH;   // 4M floats = 16MB
  float* Kbuf = (float*)d_ws;
  float* Qbuf = Kbuf + SZ;
  float* Vbuf = Qbuf + SZ;
  float* Opre = Vbuf + SZ;

  // 1) QKV projection + de-interleave: (4096x1024)x(1024x3072)
  dim3 g1(3072 / 64, 4096 / 128);
  gemm_wmma<0><<<g1, 256, 0, stream>>>(x, Wu, nullptr, Kbuf, Qbuf, Vbuf,
                                       4096, 3072, 1024);
  // 2) causal flash attention: grid = B*H * (T/128)
  attn_wmma<<<dim3(2 * H_NUM * (T_SEQ / 128)), 256, 0, stream>>>(Qbuf, Kbuf,
                                                                 Vbuf, Opre);
  // 3) output projection + bias: (4096x1024)x(1024x1024)
  dim3 g3(1024 / 64, 4096 / 128);
  gemm_wmma<1><<<g3, 256, 0, stream>>>(Opre, Wp, bp, out, nullptr, nullptr,
                                       4096, 1024, 1024);
}